// MolecularGNN_75299366633809
// MI455X (gfx1250) — compile-verified
//
#include <hip/hip_runtime.h>

#define DEMB 128

typedef float v2f __attribute__((ext_vector_type(2)));
typedef float v8f __attribute__((ext_vector_type(8)));

// ---------------------------------------------------------------- init h
__global__ __launch_bounds__(256) void init_h_kernel(
    const int* __restrict__ x, const float* __restrict__ emb_atom,
    const float* __restrict__ emb_chir, float* __restrict__ h, int N) {
  int idx = blockIdx.x * blockDim.x + threadIdx.x;   // one float4 per thread
  int total = N * (DEMB / 4);
  if (idx >= total) return;
  int row = idx >> 5, j = idx & 31;
  int a = x[2 * row], c = x[2 * row + 1];
  float4 va = ((const float4*)emb_atom)[a * 32 + j];
  float4 vc = ((const float4*)emb_chir)[c * 32 + j];
  float4 o;
  o.x = va.x + vc.x; o.y = va.y + vc.y; o.z = va.z + vc.z; o.w = va.w + vc.w;
  ((float4*)h)[idx] = o;
}

// ------------------------------------------- agg = h + ee(self-loop attrs)
__global__ __launch_bounds__(256) void agg_init_kernel(
    const float* __restrict__ h, const float* __restrict__ ee1_self,
    const float* __restrict__ ee2_self, float* __restrict__ agg, int N) {
  int idx = blockIdx.x * blockDim.x + threadIdx.x;
  int total = N * (DEMB / 4);
  if (idx >= total) return;
  int j = idx & 31;
  float4 v = ((const float4*)h)[idx];
  float4 e1 = ((const float4*)ee1_self)[j];
  float4 e2 = ((const float4*)ee2_self)[j];
  float4 o;
  o.x = v.x + e1.x + e2.x; o.y = v.y + e1.y + e2.y;
  o.z = v.z + e1.z + e2.z; o.w = v.w + e1.w + e2.w;
  ((float4*)agg)[idx] = o;
}

// ---------------------------- scatter-add messages: one wave per edge
__global__ __launch_bounds__(256) void scatter_kernel(
    const float* __restrict__ h, const int* __restrict__ src,
    const int* __restrict__ dst, const int* __restrict__ edge_attr,
    const float* __restrict__ ee1, const float* __restrict__ ee2,
    float* __restrict__ agg, int E) {
  int e = blockIdx.x * (blockDim.x >> 5) + (threadIdx.x >> 5);
  if (e >= E) return;
  int lane = threadIdx.x & 31;
  int s = src[e], d = dst[e];
  int a0 = edge_attr[2 * e], a1 = edge_attr[2 * e + 1];
  float4 v  = ((const float4*)(h + (size_t)s * DEMB))[lane];
  float4 w1 = ((const float4*)(ee1 + (size_t)a0 * DEMB))[lane];
  float4 w2 = ((const float4*)(ee2 + (size_t)a1 * DEMB))[lane];
  float* out = agg + (size_t)d * DEMB + lane * 4;
  unsafeAtomicAdd(out + 0, v.x + w1.x + w2.x);
  unsafeAtomicAdd(out + 1, v.y + w1.y + w2.y);
  unsafeAtomicAdd(out + 2, v.z + w1.z + w2.z);
  unsafeAtomicAdd(out + 3, v.w + w1.w + w2.w);
}

// ---------------------------------------------------------- WMMA GEMM
// C[M,NDIM] = relu?(A[M,KDIM] @ B[KDIM,NDIM] + bias). M % 16 == 0.
// Block = 128 threads (4 waves). Each wave owns one 16-row x 16-col tile per
// N-pass; B is staged through LDS in 32-row slabs.
template <int KDIM, int NDIM, bool RELU>
__global__ __launch_bounds__(128) void gemm_wmma_kernel(
    const float* __restrict__ A, const float* __restrict__ B,
    const float* __restrict__ bias, float* __restrict__ C, int M) {
  constexpr int NPASS = NDIM / 64;   // 4 waves * 16 cols per pass
  constexpr int KSLAB = 32;
  __shared__ float bs[KSLAB * NDIM];

  const int tid  = threadIdx.x;
  const int wave = tid >> 5;
  const int lane = tid & 31;
  const int half = lane >> 4;        // 0: K lo pair / rows 0-7, 1: K hi pair / rows 8-15
  const int l15  = lane & 15;
  const int m0   = blockIdx.x * 16;

  v8f acc[NPASS];
#pragma unroll
  for (int p = 0; p < NPASS; ++p) acc[p] = (v8f)0.0f;

  const float* Arow = A + (size_t)(m0 + l15) * KDIM;

  for (int kk = 0; kk < KDIM; kk += KSLAB) {
    // cooperative slab load B[kk .. kk+32) x NDIM  -> LDS
    const float4* Bsrc = (const float4*)(B + (size_t)kk * NDIM);
    float4* Bdst = (float4*)bs;
    constexpr int TOT4 = KSLAB * NDIM / 4;
#pragma unroll
    for (int i = tid; i < TOT4; i += 128) Bdst[i] = Bsrc[i];
    __syncthreads();

#pragma unroll
    for (int k = 0; k < KSLAB; k += 4) {
      // A fragment (16x4 f32): lanes 0-15 hold K={k,k+1}, lanes 16-31 K={k+2,k+3}
      v2f afrag = *(const v2f*)(Arow + kk + k + half * 2);
#pragma unroll
      for (int p = 0; p < NPASS; ++p) {
        int col = wave * 16 + p * 64 + l15;
        v2f bfrag;
        bfrag.x = bs[(k + half * 2 + 0) * NDIM + col];
        bfrag.y = bs[(k + half * 2 + 1) * NDIM + col];
        acc[p] = __builtin_amdgcn_wmma_f32_16x16x4_f32(
            false, afrag, false, bfrag, (short)0, acc[p], false, false);
      }
    }
    __syncthreads();
  }

  // epilogue: bias (+relu), store. C/D layout: VGPR r -> row r (lanes 0-15),
  // row r+8 (lanes 16-31); N = lane&15.
#pragma unroll
  for (int p = 0; p < NPASS; ++p) {
    int col = wave * 16 + p * 64 + l15;
    float bv = bias[col];
#pragma unroll
    for (int r = 0; r < 8; ++r) {
      int row = m0 + r + half * 8;
      float v = acc[p][r] + bv;
      if (RELU) v = fmaxf(v, 0.0f);
      C[(size_t)row * NDIM + col] = v;
    }
  }
}

// ------------------------------------------------- BN statistics helpers
__global__ void zero_sums_kernel(float* __restrict__ sums) {
  sums[threadIdx.x] = 0.0f;   // 256 threads: 128 sums + 128 sumsq
}

__global__ __launch_bounds__(128) void colsum_kernel(
    const float* __restrict__ out, float* __restrict__ sums, int M) {
  const int ROWS = 512;
  int col = threadIdx.x;
  int r0 = blockIdx.x * ROWS;
  int r1 = r0 + ROWS; if (r1 > M) r1 = M;
  float s = 0.0f, q = 0.0f;
  for (int r = r0; r < r1; ++r) {
    float v = out[(size_t)r * DEMB + col];
    s += v; q += v * v;
  }
  unsafeAtomicAdd(&sums[col], s);
  unsafeAtomicAdd(&sums[DEMB + col], q);
}

__global__ void bn_coef_kernel(const float* __restrict__ sums,
                               const float* __restrict__ g,
                               const float* __restrict__ b,
                               float* __restrict__ coef, int M) {
  int d = threadIdx.x;  // 128 threads
  float invn = 1.0f / (float)M;
  float mu = sums[d] * invn;
  float var = sums[DEMB + d] * invn - mu * mu;
  float scale = g[d] * rsqrtf(var + 1e-5f);
  coef[d] = scale;                    // scale
  coef[DEMB + d] = b[d] - mu * scale; // shift
}

__global__ __launch_bounds__(256) void bn_apply_kernel(
    const float* __restrict__ out, const float* __restrict__ coef,
    float* __restrict__ dst, int N, int do_relu) {
  int idx = blockIdx.x * blockDim.x + threadIdx.x;
  int total = N * (DEMB / 4);
  if (idx >= total) return;
  int j = idx & 31;
  float4 v = ((const float4*)out)[idx];
  float4 sc = ((const float4*)coef)[j];
  float4 sh = ((const float4*)(coef + DEMB))[j];
  float4 o;
  o.x = fmaf(v.x, sc.x, sh.x); o.y = fmaf(v.y, sc.y, sh.y);
  o.z = fmaf(v.z, sc.z, sh.z); o.w = fmaf(v.w, sc.w, sh.w);
  if (do_relu) {
    o.x = fmaxf(o.x, 0.0f); o.y = fmaxf(o.y, 0.0f);
    o.z = fmaxf(o.z, 0.0f); o.w = fmaxf(o.w, 0.0f);
  }
  ((float4*)dst)[idx] = o;
}

// --------------------------------------------------------------- launch
extern "C" void kernel_launch(void* const* d_in, const int* in_sizes, int n_in,
                              void* d_out, int out_size, void* d_ws, size_t ws_size,
                              hipStream_t stream) {
  const int*   x          = (const int*)d_in[0];
  const int*   edge_index = (const int*)d_in[1];
  const int*   edge_attr  = (const int*)d_in[2];
  const float* emb_atom   = (const float*)d_in[3];
  const float* emb_chir   = (const float*)d_in[4];
  const float* W1         = (const float*)d_in[5];
  const float* b1         = (const float*)d_in[6];
  const float* W2         = (const float*)d_in[7];
  const float* b2         = (const float*)d_in[8];
  const float* eemb1      = (const float*)d_in[9];
  const float* eemb2      = (const float*)d_in[10];
  const float* bn_w       = (const float*)d_in[11];
  const float* bn_b       = (const float*)d_in[12];

  const int N = in_sizes[0] / 2;
  const int E = in_sizes[1] / 2;
  const int L = in_sizes[5] / (DEMB * 2 * DEMB);

  // workspace carve: h | agg(=out) | hid(2x) | stats
  char* ws = (char*)d_ws;
  const size_t szh = (size_t)N * DEMB * sizeof(float);
  float* h    = (float*)(ws);
  float* agg  = (float*)(ws + szh);            // reused as `out` after GEMM2
  float* hid  = (float*)(ws + 2 * szh);        // N x 256
  float* sums = (float*)(ws + 4 * szh);        // 256 floats
  float* coef = sums + 256;                    // 256 floats

  const int elem4  = N * (DEMB / 4);
  const int gElem  = (elem4 + 255) / 256;
  const int gEdge  = (E + 7) / 8;              // 8 waves (edges) per 256-thr block
  const int gGemm  = N / 16;                   // M divisible by 16
  const int gCol   = (N + 511) / 512;

  init_h_kernel<<<gElem, 256, 0, stream>>>(x, emb_atom, emb_chir, h, N);

  for (int l = 0; l < L; ++l) {
    const float* ee1 = eemb1 + (size_t)l * 6 * DEMB;
    const float* ee2 = eemb2 + (size_t)l * 3 * DEMB;

    // self-loop (attrs [4,0]) folded into agg init
    agg_init_kernel<<<gElem, 256, 0, stream>>>(h, ee1 + 4 * DEMB, ee2, agg, N);

    scatter_kernel<<<gEdge, 256, 0, stream>>>(h, edge_index, edge_index + E,
                                              edge_attr, ee1, ee2, agg, E);

    gemm_wmma_kernel<DEMB, 2 * DEMB, true><<<gGemm, 128, 0, stream>>>(
        agg, W1 + (size_t)l * DEMB * 2 * DEMB, b1 + (size_t)l * 2 * DEMB, hid, N);

    gemm_wmma_kernel<2 * DEMB, DEMB, false><<<gGemm, 128, 0, stream>>>(
        hid, W2 + (size_t)l * 2 * DEMB * DEMB, b2 + (size_t)l * DEMB, agg, N);

    zero_sums_kernel<<<1, 256, 0, stream>>>(sums);
    colsum_kernel<<<gCol, 128, 0, stream>>>(agg, sums, N);
    bn_coef_kernel<<<1, 128, 0, stream>>>(sums, bn_w + (size_t)l * DEMB,
                                          bn_b + (size_t)l * DEMB, coef, N);

    float* dst = (l == L - 1) ? (float*)d_out : h;
    bn_apply_kernel<<<gElem, 256, 0, stream>>>(agg, coef, dst, N, l < L - 1);
  }
}